// FFM_79826262163465
// MI455X (gfx1250) — compile-verified
//
#include <hip/hip_runtime.h>
#include <math.h>

#define N_DENSE   13
#define FIELD_NUM 39
#define FEATURE_NUM 260013
#define KDIM 16
#define BATCH 4096
#define THREADS 256
#define NWAVES (THREADS / 32)

typedef __attribute__((ext_vector_type(2))) float v2f;
typedef __attribute__((ext_vector_type(8))) float v8f;

__global__ __launch_bounds__(THREADS) void ffm_kernel(
    const float* __restrict__ x,        // [B, 39]
    const float* __restrict__ W,        // [39, 260013, 16]
    const float* __restrict__ fc_w,     // [260013]
    const float* __restrict__ bias,     // [1]
    const int*   __restrict__ offsets,  // [39]
    float*       __restrict__ out)      // [B]
{
    __shared__ int   s_xf[FIELD_NUM];
    __shared__ float s_xm[FIELD_NUM];
    __shared__ float s_red;

    const int b   = blockIdx.x;
    const int tid = threadIdx.x;
    const int wave = tid >> 5;
    const int L    = tid & 31;

    if (tid == 0) s_red = 0.0f;
    __syncthreads();

    // ---- Stage 0: feature indices, multipliers, linear term ----
    if (tid < FIELD_NUM) {
        float xv = x[(size_t)b * FIELD_NUM + tid];
        int   base;
        float xm;
        if (tid < N_DENSE) { base = 0; xm = xv; }
        else               { base = (int)floorf(xv); xm = 1.0f; }
        int f = base + offsets[tid];
        s_xf[tid] = f;
        s_xm[tid] = xm;
        atomicAdd(&s_red, fc_w[f]);     // linear term partial
    }
    __syncthreads();

    // ---- Stage 1: upper-triangular (i, j-chunk) tiles via WMMA diag trick ----
    const int m  = L & 15;
    const int hi = L >> 4;
    float accl = 0.0f;

    int t = 0;
    for (int i = 0; i < FIELD_NUM - 1; ++i) {                 // rows i < 38
        const int   feat_i = s_xf[i];
        const float xm_i   = s_xm[i];
        const float* __restrict__ tblBase =
            W + (size_t)feat_i * KDIM;                        // + field*FEATURE_NUM*K later
        for (int jc = (i >> 4); jc < 3; ++jc, ++t) {
            if ((t & (NWAVES - 1)) != wave) continue;

            const int j0 = jc << 4;
            const int jA = j0 + m;
            const bool act = (jA >= i) && (jA <= FIELD_NUM - 1);
            const int jcl = act ? jA : (FIELD_NUM - 1);       // clamped field/col idx

            // A row m: W[i, xf[jA], :]   (this lane holds k = 4s+2*hi .. +1)
            const float* __restrict__ rowA =
                W + ((size_t)i * FEATURE_NUM + (size_t)s_xf[jcl]) * KDIM;
            // B col n(=m): W[field jA, xf[i], :]
            const float* __restrict__ rowB =
                tblBase + (size_t)jcl * ((size_t)FEATURE_NUM * KDIM);

            // Zero fragments once, then one predicated region with all 8 loads
            v2f a0 = {0.f,0.f}, a1 = {0.f,0.f}, a2 = {0.f,0.f}, a3 = {0.f,0.f};
            v2f b0 = {0.f,0.f}, b1 = {0.f,0.f}, b2 = {0.f,0.f}, b3 = {0.f,0.f};
            if (act) {
                const float* pa = rowA + 2 * hi;
                a0 = *(const v2f*)(pa +  0);
                a1 = *(const v2f*)(pa +  4);
                a2 = *(const v2f*)(pa +  8);
                a3 = *(const v2f*)(pa + 12);
                float4 q0 = *(const float4*)(rowB +  0);
                float4 q1 = *(const float4*)(rowB +  4);
                float4 q2 = *(const float4*)(rowB +  8);
                float4 q3 = *(const float4*)(rowB + 12);
                // B fragment slab s: VGPR0 = B[4s+hi, n], VGPR1 = B[4s+hi+2, n]
                b0.x = hi ? q0.y : q0.x;  b0.y = hi ? q0.w : q0.z;
                b1.x = hi ? q1.y : q1.x;  b1.y = hi ? q1.w : q1.z;
                b2.x = hi ? q2.y : q2.x;  b2.y = hi ? q2.w : q2.z;
                b3.x = hi ? q3.y : q3.x;  b3.y = hi ? q3.w : q3.z;
            }

            v8f c;
            c = __builtin_amdgcn_wmma_f32_16x16x4_f32(
                    false, a0, false, b0, (short)0,
                    (v8f){0.f,0.f,0.f,0.f,0.f,0.f,0.f,0.f}, false, false);
            c = __builtin_amdgcn_wmma_f32_16x16x4_f32(
                    false, a1, false, b1, (short)0, c, false, false);
            c = __builtin_amdgcn_wmma_f32_16x16x4_f32(
                    false, a2, false, b2, (short)0, c, false, false);
            c = __builtin_amdgcn_wmma_f32_16x16x4_f32(
                    false, a3, false, b3, (short)0, c, false, false);

            // weight for this lane's diagonal element (if any)
            const int m_d = (L < 8) ? L : ((L >= 24) ? (L - 16) : -1);
            float wt = 0.0f;
            if (m_d >= 0) {
                int jD = j0 + m_d;
                if (jD >= i && jD <= FIELD_NUM - 1) {
                    float cf = (jD == i || jD == FIELD_NUM - 1) ? 1.0f : 2.0f;
                    wt = cf * xm_i * s_xm[jD];
                }
            }
            // extract diag(C): VGPR r -> lane r (M=N=r), lane r+24 (M=N=r+8)
            float diag = 0.0f;
            #pragma unroll
            for (int r = 0; r < 8; ++r) {
                float v = c[r];
                diag += ((L == r) || (L == r + 24)) ? v : 0.0f;
            }
            accl += diag * wt;
        }
    }

    // ---- Stage 2: reduce + sigmoid ----
    atomicAdd(&s_red, accl);
    __syncthreads();
    if (tid == 0) {
        float z = s_red + bias[0];
        out[b] = 1.0f / (1.0f + expf(-z));
    }
}

extern "C" void kernel_launch(void* const* d_in, const int* in_sizes, int n_in,
                              void* d_out, int out_size, void* d_ws, size_t ws_size,
                              hipStream_t stream) {
    const float* x       = (const float*)d_in[0];
    const float* W       = (const float*)d_in[1];
    const float* fc_w    = (const float*)d_in[2];
    const float* bias    = (const float*)d_in[3];
    const int*   offsets = (const int*)d_in[4];
    float* out = (float*)d_out;
    ffm_kernel<<<BATCH, THREADS, 0, stream>>>(x, W, fc_w, bias, offsets, out);
}